// AstroGCNLayer_22342419874159
// MI455X (gfx1250) — compile-verified
//
#include <hip/hip_runtime.h>

// AstroGCNLayer on MI455X (gfx1250).
// - fp32 WMMA (V_WMMA_F32_16X16X4_F32) for all GEMMs (reference-accurate fp32).
// - Deterministic counting sort of edges by type: each 128-edge tile is
//   (almost always) type-homogeneous, so the type-dependent GEMM runs one
//   weight band instead of four.  Mixed boundary tiles handled by per-lane
//   A-base select against a zeroed LDS row (no per-element masking VALU).
// - Weights repacked once per launch into B-operand order in d_ws: each lane
//   fetches B with one coalesced global_load_b64 (wave reads contiguous 256B).
// - 128 edges (8 x 16-row sub-tiles) per block: one B load feeds 8 WMMAs.
// - fp32 global atomics for the scatter-add; LayerNorm+ReLU fused node kernel.

#define D 128
#define LSTRIDE 132        // padded LDS row stride (floats), conflict-free ds_load_b64
#define EROWS 128          // edges per block
#define ESUB 8             // 16-row sub-tiles per wave (edge kernel)
#define NROWS 64           // nodes per block
#define NSUB 4
#define ZROW (EROWS * LSTRIDE)   // offset of the zeroed LDS row

typedef float v2f __attribute__((ext_vector_type(2)));
typedef float v8f __attribute__((ext_vector_type(8)));

#define WP_MAT (D * D)     // floats per repacked matrix
#define WP_E1  (4 * WP_MAT)
#define WP_E2  (5 * WP_MAT)
#define WP_SELF (6 * WP_MAT)

__device__ __forceinline__ v8f wmma_f32x4(v2f a, v2f b, v8f c) {
    return __builtin_amdgcn_wmma_f32_16x16x4_f32(
        false, a, false, b, (short)0, c, false, false);
}

__global__ __launch_bounds__(256) void gcn_zero_kernel(float* __restrict__ p, int n) {
    int i = blockIdx.x * 256 + threadIdx.x;
    if (i < n) p[i] = 0.0f;
}

// Repack DxD row-major matrices into B-operand order:
// Wp[m][kb/4][n][j] = W[m][kb/4*4 + j][n]   (j = 0..3)
__global__ __launch_bounds__(256) void gcn_repack_kernel(
    const float* __restrict__ W, float* __restrict__ Wp, int total) {
    int i = blockIdx.x * 256 + threadIdx.x;
    if (i >= total) return;
    int m   = i >> 14;
    int rem = i & (WP_MAT - 1);
    int kbi = rem >> 9;
    int nj  = rem & 511;
    Wp[i] = W[m * WP_MAT + (kbi * 4 + (nj & 3)) * D + (nj >> 2)];
}

// ---- deterministic counting sort of edges by type (T = 4) ----
__global__ __launch_bounds__(256) void gcn_count_kernel(
    const int* __restrict__ edge_type, int* __restrict__ counts, int E) {
    __shared__ int c4[4];
    if (threadIdx.x < 4) c4[threadIdx.x] = 0;
    __syncthreads();
    int e = blockIdx.x * 256 + threadIdx.x;
    if (e < E) atomicAdd(&c4[edge_type[e]], 1);
    __syncthreads();
    if (threadIdx.x < 4) counts[blockIdx.x * 4 + threadIdx.x] = c4[threadIdx.x];
}

__global__ __launch_bounds__(256) void gcn_scan_kernel(
    int* __restrict__ counts, int* __restrict__ bases, int nchunks) {
    __shared__ int totals[4];
    int t = threadIdx.x;
    if (t < 4) {
        int run = 0;
        for (int c = 0; c < nchunks; ++c) {
            int v = counts[c * 4 + t];
            counts[c * 4 + t] = run;     // exclusive prefix within type
            run += v;
        }
        totals[t] = run;
    }
    __syncthreads();
    if (t == 0) {
        int b = 0;
        for (int q = 0; q < 4; ++q) { bases[q] = b; b += totals[q]; }
    }
    __syncthreads();
    if (t < 4) {
        int b = bases[t];
        for (int c = 0; c < nchunks; ++c) counts[c * 4 + t] += b;
    }
}

__global__ __launch_bounds__(256) void gcn_permute_kernel(
    const int* __restrict__ edge_type, const int* __restrict__ counts,
    int* __restrict__ perm, int E) {
    __shared__ int lt[256];
    int e = blockIdx.x * 256 + threadIdx.x;
    int myt = (e < E) ? edge_type[e] : -1;
    lt[threadIdx.x] = myt;
    __syncthreads();
    if (e < E) {
        int rank = 0;
        for (int j = 0; j < (int)threadIdx.x; ++j) rank += (lt[j] == myt);
        perm[counts[blockIdx.x * 4 + myt] + rank] = e;   // deterministic position
    }
}

// ---- main edge kernel: one block = 128 type-sorted edges ----
__global__ __launch_bounds__(256) void gcn_edge_kernel(
    const float* __restrict__ x,          // [N, D]
    const int*   __restrict__ edge_index, // [2, E] (row=dst, col=src)
    const int*   __restrict__ edge_type,  // [E]
    const float* __restrict__ ef,         // [E, D]
    const float* __restrict__ b_types,    // [T, D]
    const float* __restrict__ b_e1,
    const float* __restrict__ b_e2,
    const float* __restrict__ wp,         // repacked weights
    const int*   __restrict__ perm,       // type-sorted edge order
    float* __restrict__ out,              // [N, D] accumulator
    int E)
{
    __shared__ __align__(16) float lds_x[(EROWS + 1) * LSTRIDE]; // + zero row
    __shared__ __align__(16) float lds_f[EROWS * LSTRIDE];       // ef, then hidden

    const int e0   = blockIdx.x * EROWS;
    const int tid  = threadIdx.x;
    const int lane = tid & 31;
    const int wave = tid >> 5;
    const int n0   = wave * 16;
    const int rowA = lane & 15;
    const int hi   = lane >> 4;
    const int koff = 2 * hi;
    const int nl   = n0 + (lane & 15);

    for (int i = tid; i < LSTRIDE; i += 256) lds_x[ZROW + i] = 0.0f;
    for (int i = tid; i < EROWS * (D / 4); i += 256) {
        int r = i >> 5;
        int c = (i & 31) << 2;
        int pos = e0 + r; if (pos >= E) pos = E - 1;
        int e = perm[pos];
        int src = edge_index[E + e];       // col (source node)
        *(float4*)&lds_x[r * LSTRIDE + c] = *(const float4*)&x[src * D + c];
        *(float4*)&lds_f[r * LSTRIDE + c] = *(const float4*)&ef[e * D + c];
    }
    __syncthreads();

    const v8f vz = {0.f,0.f,0.f,0.f,0.f,0.f,0.f,0.f};

    // ---- hidden = relu(ef @ W_e1 + b_e1) ----
    v8f acc[ESUB];
    #pragma unroll
    for (int s = 0; s < ESUB; ++s) acc[s] = vz;
    {
        const float* __restrict__ wpe1 = wp + WP_E1;
        for (int kbi = 0; kbi < D / 4; ++kbi) {
            v2f b = *(const v2f*)&wpe1[(kbi * D + nl) * 4 + koff];
            int ak = kbi * 4 + koff;
            #pragma unroll
            for (int s = 0; s < ESUB; ++s) {
                v2f a = *(const v2f*)&lds_f[(s * 16 + rowA) * LSTRIDE + ak];
                acc[s] = wmma_f32x4(a, b, acc[s]);
            }
        }
    }
    __syncthreads();
    {
        float be1 = b_e1[nl];
        #pragma unroll
        for (int s = 0; s < ESUB; ++s)
            #pragma unroll
            for (int v = 0; v < 8; ++v) {
                float h = acc[s][v] + be1;
                lds_f[(s * 16 + v + 8 * hi) * LSTRIDE + nl] = h > 0.0f ? h : 0.0f;
            }
    }
    __syncthreads();

    // ---- msgs = hidden @ W_e2 + x_col @ W_types[type] ----
    v8f o[ESUB];
    #pragma unroll
    for (int s = 0; s < ESUB; ++s) o[s] = vz;
    {
        const float* __restrict__ wpe2 = wp + WP_E2;
        for (int kbi = 0; kbi < D / 4; ++kbi) {
            v2f b = *(const v2f*)&wpe2[(kbi * D + nl) * 4 + koff];
            int ak = kbi * 4 + koff;
            #pragma unroll
            for (int s = 0; s < ESUB; ++s) {
                v2f a = *(const v2f*)&lds_f[(s * 16 + rowA) * LSTRIDE + ak];
                o[s] = wmma_f32x4(a, b, o[s]);
            }
        }
    }
    {
        int my_t[ESUB];
        #pragma unroll
        for (int s = 0; s < ESUB; ++s) {
            int pos = e0 + s * 16 + rowA; if (pos >= E) pos = E - 1;
            my_t[s] = edge_type[perm[pos]];
        }
        int lastpos = e0 + EROWS - 1; if (lastpos >= E) lastpos = E - 1;
        const int tmin = edge_type[perm[e0]];          // sorted: block-uniform band
        const int tmax = edge_type[perm[lastpos]];
        for (int t = tmin; t <= tmax; ++t) {           // ~1 iteration after sorting
            const float* __restrict__ wpt = wp + t * WP_MAT;
            int ab[ESUB];                              // A base: real row or zero row
            #pragma unroll
            for (int s = 0; s < ESUB; ++s)
                ab[s] = (my_t[s] == t) ? (s * 16 + rowA) * LSTRIDE : ZROW;
            for (int kbi = 0; kbi < D / 4; ++kbi) {
                v2f b = *(const v2f*)&wpt[(kbi * D + nl) * 4 + koff];
                int ak = kbi * 4 + koff;
                #pragma unroll
                for (int s = 0; s < ESUB; ++s) {
                    v2f a = *(const v2f*)&lds_x[ab[s] + ak];
                    o[s] = wmma_f32x4(a, b, o[s]);
                }
            }
        }
    }

    // ---- biases + scatter-add to destination nodes ----
    const float be2 = b_e2[nl];
    #pragma unroll
    for (int s = 0; s < ESUB; ++s)
        #pragma unroll
        for (int v = 0; v < 8; ++v) {
            int pos = e0 + s * 16 + v + 8 * hi;
            if (pos < E) {
                int e   = perm[pos];
                int t   = edge_type[e];
                int dst = edge_index[e];
                float val = o[s][v] + be2 + b_types[t * D + nl];
                atomicAdd(&out[dst * D + nl], val);
            }
        }
}

// One block = 64 nodes: out = relu(LN(accum + x @ W_self + b_self))
__global__ __launch_bounds__(256) void gcn_node_kernel(
    const float* __restrict__ x,
    const float* __restrict__ b_self,
    const float* __restrict__ ln_g, const float* __restrict__ ln_b,
    const float* __restrict__ wp,
    float* __restrict__ out, int N)
{
    __shared__ __align__(16) float lds_x[NROWS * LSTRIDE];
    __shared__ __align__(16) float lds_r[NROWS * LSTRIDE];
    __shared__ float mu_s[NROWS];
    __shared__ float rs_s[NROWS];

    const int v0   = blockIdx.x * NROWS;
    const int tid  = threadIdx.x;
    const int lane = tid & 31;
    const int wave = tid >> 5;
    const int n0   = wave * 16;
    const int rowA = lane & 15;
    const int hi   = lane >> 4;
    const int koff = 2 * hi;
    const int nl   = n0 + (lane & 15);

    for (int i = tid; i < NROWS * (D / 4); i += 256) {
        int r = i >> 5;
        int c = (i & 31) << 2;
        int node = v0 + r; if (node >= N) node = N - 1;
        *(float4*)&lds_x[r * LSTRIDE + c] = *(const float4*)&x[node * D + c];
    }
    __syncthreads();

    const v8f vz = {0.f,0.f,0.f,0.f,0.f,0.f,0.f,0.f};
    v8f acc[NSUB];
    #pragma unroll
    for (int s = 0; s < NSUB; ++s) acc[s] = vz;
    {
        const float* __restrict__ wps = wp + WP_SELF;
        for (int kbi = 0; kbi < D / 4; ++kbi) {
            v2f b = *(const v2f*)&wps[(kbi * D + nl) * 4 + koff];
            int ak = kbi * 4 + koff;
            #pragma unroll
            for (int s = 0; s < NSUB; ++s) {
                v2f a = *(const v2f*)&lds_x[(s * 16 + rowA) * LSTRIDE + ak];
                acc[s] = wmma_f32x4(a, b, acc[s]);
            }
        }
    }
    {
        const float bs = b_self[nl];
        #pragma unroll
        for (int s = 0; s < NSUB; ++s)
            #pragma unroll
            for (int v = 0; v < 8; ++v) {
                int r = s * 16 + v + 8 * hi;
                int node = v0 + r; if (node >= N) node = N - 1;
                lds_r[r * LSTRIDE + nl] = acc[s][v] + bs + out[node * D + nl];
            }
    }
    __syncthreads();

    {   // LayerNorm stats: 4 threads per row, shfl_xor reduce within 4 lanes
        int r = tid >> 2, q = tid & 3;
        const float* rp = &lds_r[r * LSTRIDE + q * 32];
        float s = 0.0f;
        for (int c = 0; c < 32; ++c) s += rp[c];
        s += __shfl_xor(s, 1); s += __shfl_xor(s, 2);
        float mu = s * (1.0f / D);
        float vs = 0.0f;
        for (int c = 0; c < 32; ++c) { float d = rp[c] - mu; vs += d * d; }
        vs += __shfl_xor(vs, 1); vs += __shfl_xor(vs, 2);
        if (q == 0) {
            mu_s[r] = mu;
            rs_s[r] = rsqrtf(vs * (1.0f / D) + 1e-5f);
        }
    }
    __syncthreads();

    for (int i = tid; i < NROWS * D; i += 256) {
        int r = i >> 7, c = i & (D - 1);
        int node = v0 + r;
        if (node < N) {
            float val = (lds_r[r * LSTRIDE + c] - mu_s[r]) * rs_s[r] * ln_g[c] + ln_b[c];
            out[node * D + c] = val > 0.0f ? val : 0.0f;
        }
    }
}

extern "C" void kernel_launch(void* const* d_in, const int* in_sizes, int n_in,
                              void* d_out, int out_size, void* d_ws, size_t ws_size,
                              hipStream_t stream) {
    const float* x          = (const float*)d_in[0];
    const int*   edge_index = (const int*)  d_in[1];
    const int*   edge_type  = (const int*)  d_in[2];
    const float* ef         = (const float*)d_in[3];
    const float* W_types    = (const float*)d_in[4];
    const float* b_types    = (const float*)d_in[5];
    const float* W_self     = (const float*)d_in[6];
    const float* b_self     = (const float*)d_in[7];
    const float* W_e1       = (const float*)d_in[8];
    const float* b_e1       = (const float*)d_in[9];
    const float* W_e2       = (const float*)d_in[10];
    const float* b_e2       = (const float*)d_in[11];
    const float* ln_g       = (const float*)d_in[12];
    const float* ln_b       = (const float*)d_in[13];
    float* out = (float*)d_out;

    const int E = in_sizes[2];
    const int N = in_sizes[0] / D;
    const int nchunks = (E + 255) / 256;

    // workspace layout
    char* wsb = (char*)d_ws;
    float* wp     = (float*)wsb;                                   // 7*D*D floats
    size_t off    = (size_t)7 * WP_MAT * sizeof(float);
    int*   counts = (int*)(wsb + off); off += (size_t)nchunks * 4 * sizeof(int);
    int*   bases  = (int*)(wsb + off); off += 16;
    int*   perm   = (int*)(wsb + off);                             // E ints

    // repack weights into B-operand order (tiny, once per launch)
    gcn_repack_kernel<<<(4 * WP_MAT + 255) / 256, 256, 0, stream>>>(W_types, wp, 4 * WP_MAT);
    gcn_repack_kernel<<<(WP_MAT + 255) / 256, 256, 0, stream>>>(W_e1,  wp + WP_E1,  WP_MAT);
    gcn_repack_kernel<<<(WP_MAT + 255) / 256, 256, 0, stream>>>(W_e2,  wp + WP_E2,  WP_MAT);
    gcn_repack_kernel<<<(WP_MAT + 255) / 256, 256, 0, stream>>>(W_self, wp + WP_SELF, WP_MAT);

    // deterministic counting sort of edges by type
    gcn_count_kernel<<<nchunks, 256, 0, stream>>>(edge_type, counts, E);
    gcn_scan_kernel<<<1, 256, 0, stream>>>(counts, bases, nchunks);
    gcn_permute_kernel<<<nchunks, 256, 0, stream>>>(edge_type, counts, perm, E);

    gcn_zero_kernel<<<(N * D + 255) / 256, 256, 0, stream>>>(out, N * D);

    gcn_edge_kernel<<<(E + EROWS - 1) / EROWS, 256, 0, stream>>>(
        x, edge_index, edge_type, ef, b_types, b_e1, b_e2, wp, perm, out, E);

    gcn_node_kernel<<<(N + NROWS - 1) / NROWS, 256, 0, stream>>>(
        x, b_self, ln_g, ln_b, wp, out, N);
}